// Memory_7344394076626
// MI455X (gfx1250) — compile-verified
//
#include <hip/hip_runtime.h>
#include <hip/hip_bf16.h>

#define MEM      262144
#define KD       256
#define BATCH    256
#define TOPK     256
#define CAND_CAP 4096
#define TPB      8      // key-tiles (of 16 slots) per block in scores kernel

typedef float v2f __attribute__((ext_vector_type(2)));
typedef float v8f __attribute__((ext_vector_type(8)));
typedef int   v4i __attribute__((ext_vector_type(4)));

#define AS1 __attribute__((address_space(1)))
#define AS3 __attribute__((address_space(3)))

// ---------- monotonic float <-> uint mapping (ascending) ----------
__device__ __forceinline__ unsigned mapf(float f) {
    unsigned b = __float_as_uint(f);
    return (b & 0x80000000u) ? ~b : (b | 0x80000000u);
}
__device__ __forceinline__ float unmapf(unsigned u) {
    unsigned b = (u & 0x80000000u) ? (u & 0x7FFFFFFFu) : ~u;
    return __uint_as_float(b);
}

// ---------- CDNA5 async copy global->LDS helpers ----------
__device__ __forceinline__ void async_load_b128(const float* g, float* l) {
#if __has_builtin(__builtin_amdgcn_global_load_async_to_lds_b128)
    __builtin_amdgcn_global_load_async_to_lds_b128((AS1 v4i*)g, (AS3 v4i*)l, 0, 0);
#else
    unsigned loff = (unsigned)(uintptr_t)l;   // LDS addr = low 32 bits of generic ptr
    asm volatile("global_load_async_to_lds_b128 %0, %1, off"
                 :: "v"(loff), "v"(g) : "memory");
#endif
}
__device__ __forceinline__ void wait_async0() {
#if __has_builtin(__builtin_amdgcn_s_wait_asynccnt)
    __builtin_amdgcn_s_wait_asynccnt(0);
#else
    asm volatile("s_wait_asynccnt 0x0" ::: "memory");
#endif
}

// ---------- K0: query = normalize(x @ W^T + b) ----------
__global__ __launch_bounds__(256) void query_kernel(const float* __restrict__ x,
                                                    const float* __restrict__ W,
                                                    const float* __restrict__ b,
                                                    float* __restrict__ q) {
    __shared__ float xr[KD];
    __shared__ float red[KD];
    int i = blockIdx.x, j = threadIdx.x;
    xr[j] = x[i * KD + j];
    __syncthreads();
    float acc = b[j];
    const float* w = W + j * KD;
    for (int k = 0; k < KD; ++k) acc += xr[k] * w[k];
    red[j] = acc * acc;
    __syncthreads();
    for (int s = 128; s > 0; s >>= 1) { if (j < s) red[j] += red[j + s]; __syncthreads(); }
    float inv = 1.0f / fmaxf(sqrtf(red[0]), 1e-12f);
    q[i * KD + j] = acc * inv;
}

// ---------- K1: scores = query @ keys^T via V_WMMA_F32_16X16X4_F32 ----------
// 512 threads = 16 wave32; each block processes TPB consecutive 16-slot key
// tiles with double-buffered LDS filled by GLOBAL_LOAD_ASYNC_TO_LDS_B128
// (ASYNCcnt), overlapping the next tile's HBM fetch with the current WMMAs.
// Each wave caches its whole 16-row A slice (64 x v2f = 128 VGPRs) once.
__global__ __launch_bounds__(512) void scores_kernel(const float* __restrict__ q,
                                                     const float* __restrict__ keys,
                                                     float* __restrict__ scores) {
    __shared__ float Bt[2][16 * KD];   // two 16 KB tile buffers
    int tileBase = blockIdx.x * TPB;
    int tid  = threadIdx.x;
    int wave = tid >> 5;               // 0..15 (wave32)
    int lane = tid & 31;
    int rowBase = wave * 16;
    int m  = lane & 15;                // A row (M) / B column (N)
    int kp = lane >> 4;                // K pair select

    // cache A slice in registers: a[kk] = query[rowBase+m][kk*4 + kp*2 .. +1]
    v2f areg[KD / 4];
    {
        const float* qrow = q + (size_t)(rowBase + m) * KD;
#pragma unroll
        for (int kk = 0; kk < KD / 4; ++kk) {
            int k0 = kk * 4 + kp * 2;
            areg[kk].x = qrow[k0];
            areg[kk].y = qrow[k0 + 1];
        }
    }

    // prologue: async fill buffer 0 with tile 0 (2 x B128 per thread = 16 KB)
    {
        const float* gsrc = keys + (size_t)tileBase * 16 * KD;
#pragma unroll
        for (int k = 0; k < 2; ++k) {
            int i4 = tid + k * 512;                 // float4 index
            async_load_b128(gsrc + i4 * 4, &Bt[0][i4 * 4]);
        }
    }

    for (int t = 0; t < TPB; ++t) {
        int cur = t & 1;
        wait_async0();          // this wave's fills of Bt[cur] landed
        __syncthreads();        // all waves' fills landed; prev readers done
        if (t + 1 < TPB) {
            const float* gsrc = keys + (size_t)(tileBase + t + 1) * 16 * KD;
#pragma unroll
            for (int k = 0; k < 2; ++k) {
                int i4 = tid + k * 512;
                async_load_b128(gsrc + i4 * 4, &Bt[1 - cur][i4 * 4]);
            }
        }
        const float* brow = &Bt[cur][m * KD];
        v8f c = {};
#pragma unroll
        for (int kk = 0; kk < KD / 4; ++kk) {
            int k0 = kk * 4 + kp * 2;
            v2f bb;
            bb.x = brow[k0];
            bb.y = brow[k0 + 1];
            c = __builtin_amdgcn_wmma_f32_16x16x4_f32(false, areg[kk], false, bb,
                                                      (short)0, c, false, false);
        }
        // C layout: VGPR r -> lanes 0-15: M=r, lanes 16-31: M=r+8; N = lane%16
        int slotBase = (tileBase + t) * 16;
        int n = lane & 15;
        int rOff = (lane >> 4) * 8;
        for (int r = 0; r < 8; ++r)
            scores[(size_t)(rowBase + r + rOff) * MEM + slotBase + n] = c[r];
        __syncthreads();        // Bt[cur] free for refill next iteration
    }
}

// ---------- K2: per-row 12-bit histogram of mapped scores ----------
__global__ __launch_bounds__(1024) void hist12_kernel(const float* __restrict__ scores,
                                                      unsigned* __restrict__ hist) {
    __shared__ unsigned h[4096];
    int row = blockIdx.x;
    for (int i = threadIdx.x; i < 4096; i += 1024) h[i] = 0;
    __syncthreads();
    const float* s = scores + (size_t)row * MEM;
    for (int i = threadIdx.x; i < MEM; i += 1024)
        atomicAdd(&h[mapf(s[i]) >> 20], 1u);
    __syncthreads();
    for (int i = threadIdx.x; i < 4096; i += 1024) hist[row * 4096 + i] = h[i];
}

// ---------- K3: per-row threshold bin (top-256 boundary) ----------
__global__ void thresh_kernel(const unsigned* __restrict__ hist,
                              unsigned* __restrict__ thresh) {
    int row = threadIdx.x;
    const unsigned* h = hist + row * 4096;
    unsigned cum = 0, t = 0;
    for (int b = 4095; b >= 0; --b) {
        cum += h[b];
        if (cum >= TOPK) { t = (unsigned)b; break; }
    }
    thresh[row] = t;
}

// ---------- K4: per-row candidate gather (bin >= threshold) ----------
__global__ __launch_bounds__(1024) void gather_kernel(const float* __restrict__ scores,
                                                      const unsigned* __restrict__ thresh,
                                                      unsigned long long* __restrict__ cand,
                                                      unsigned* __restrict__ cnt) {
    int row = blockIdx.x;
    unsigned t = thresh[row];
    const float* s = scores + (size_t)row * MEM;
    unsigned long long* crow = cand + (size_t)row * CAND_CAP;
    for (int i = threadIdx.x; i < MEM; i += 1024) {
        unsigned u = mapf(s[i]);
        if ((u >> 20) >= t) {
            unsigned p = atomicAdd(&cnt[row], 1u);
            if (p < CAND_CAP)
                crow[p] = ((unsigned long long)(~u) << 32) | (unsigned)i;  // asc key = desc score, asc idx
        }
    }
}

// ---------- shared bitonic sort of 4096 u64 keys (ascending) ----------
__device__ __forceinline__ void bitonic4096(unsigned long long* s) {
    for (int k = 2; k <= 4096; k <<= 1)
        for (int j = k >> 1; j > 0; j >>= 1) {
            for (int idx = threadIdx.x; idx < 4096; idx += blockDim.x) {
                int p = idx ^ j;
                if (p > idx) {
                    bool up = ((idx & k) == 0);
                    unsigned long long A = s[idx], B = s[p];
                    if ((A > B) == up) { s[idx] = B; s[p] = A; }
                }
            }
            __syncthreads();
        }
}

// ---------- K5: per-row sort, softmax, margin-loss pieces ----------
__global__ __launch_bounds__(1024) void topk_kernel(const unsigned long long* __restrict__ cand,
                                                    const unsigned* __restrict__ cnt,
                                                    const int* __restrict__ values,
                                                    const int* __restrict__ y,
                                                    float* __restrict__ out_soft,
                                                    int* __restrict__ out_yhat,
                                                    float* __restrict__ lossarr,
                                                    unsigned* __restrict__ yhatidx,
                                                    int* __restrict__ correctw) {
    __shared__ unsigned long long s[4096];
    __shared__ float cosv[TOPK];
    __shared__ int   idxv[TOPK];
    __shared__ float rA[TOPK], rB[TOPK], rC[TOPK], rE[TOPK];
    int row = blockIdx.x, tid = threadIdx.x;
    unsigned n = cnt[row]; if (n > CAND_CAP) n = CAND_CAP;
    const unsigned long long* crow = cand + (size_t)row * CAND_CAP;
    for (int i = tid; i < 4096; i += 1024)
        s[i] = (i < (int)n) ? crow[i] : 0xFFFFFFFFFFFFFFFFull;
    __syncthreads();
    bitonic4096(s);

    if (tid < TOPK) {
        unsigned long long key = s[tid];
        unsigned u = ~(unsigned)(key >> 32);
        cosv[tid] = unmapf(u);
        idxv[tid] = (int)(unsigned)(key & 0xFFFFFFFFu);
    }
    __syncthreads();

    int yr = y[row];
    if (tid < TOPK) {
        int tv = values[idxv[tid]];
        float maskf = (tv == yr) ? 1.0f : 0.0f;
        rA[tid] = cosv[tid] * maskf;            // pos candidates (zeros included)
        rB[tid] = cosv[tid] * (1.0f - maskf);   // neg candidates
        rC[tid] = maskf;                        // has_pos count
        rE[tid] = expf(cosv[tid] - cosv[0]);    // TEMP == 1.0
    }
    __syncthreads();
    for (int st = 128; st > 0; st >>= 1) {
        if (tid < st) {
            rA[tid] = fmaxf(rA[tid], rA[tid + st]);
            rB[tid] = fmaxf(rB[tid], rB[tid + st]);
            rC[tid] += rC[tid + st];
            rE[tid] += rE[tid + st];
        }
        __syncthreads();
    }
    if (tid < TOPK)
        out_soft[row * TOPK + tid] = expf(cosv[tid] - cosv[0]) / rE[0];
    if (tid == 0) {
        int yh_idx = idxv[0];
        int yh = values[yh_idx];
        out_yhat[row] = yh;
        yhatidx[row] = (unsigned)yh_idx;
        correctw[row] = (yh == yr) ? 1 : 0;
        float haspos = (rC[0] > 0.0f) ? 1.0f : 0.0f;
        float pos = rA[0] * haspos;
        lossarr[row] = fmaxf(rB[0] - pos + 0.1f, 0.0f);
    }
}

// ---------- K6: keys copy (overwrites score scratch region) ----------
__global__ void copy_keys(const float* __restrict__ keys, float* __restrict__ out_keys) {
    size_t i = (size_t)blockIdx.x * blockDim.x + threadIdx.x;
    size_t stride = (size_t)gridDim.x * blockDim.x;
    size_t total = (size_t)MEM * KD;
    for (; i < total; i += stride) out_keys[i] = keys[i];
}

// ---------- K7: values copy, age += 1 ----------
__global__ void copy_vals_age(const int* __restrict__ vals, const float* __restrict__ age,
                              int* __restrict__ out_vals, float* __restrict__ out_age) {
    int i = blockIdx.x * blockDim.x + threadIdx.x;
    if (i < MEM) { out_vals[i] = vals[i]; out_age[i] = age[i] + 1.0f; }
}

// ---------- K8: correct-row scatter (serialized, jax last-wins order) ----------
__global__ __launch_bounds__(256) void correct_update(const float* __restrict__ keys_in,
                                                      const float* __restrict__ q,
                                                      const unsigned* __restrict__ yhatidx,
                                                      const int* __restrict__ correctw,
                                                      float* __restrict__ out_keys,
                                                      float* __restrict__ out_age) {
    __shared__ float red[256];
    int tid = threadIdx.x;
    for (int r = 0; r < BATCH; ++r) {
        if (!correctw[r]) continue;
        unsigned d = yhatidx[r];
        float v = keys_in[(size_t)d * KD + tid] + q[r * KD + tid];
        red[tid] = v * v;
        __syncthreads();
        for (int s = 128; s > 0; s >>= 1) { if (tid < s) red[tid] += red[tid + s]; __syncthreads(); }
        float inv = 1.0f / fmaxf(sqrtf(red[0]), 1e-12f);
        __syncthreads();
        out_keys[(size_t)d * KD + tid] = v * inv;
        if (tid == 0) out_age[d] = 0.0f;
        __syncthreads();
    }
}

// ---------- K9: 16-bit global histogram of age_noisy ----------
__global__ void age_hist(const float* __restrict__ out_age, const float* __restrict__ un,
                         unsigned* __restrict__ hist16) {
    int i = blockIdx.x * blockDim.x + threadIdx.x;
    if (i >= MEM) return;
    float an = out_age[i] + (2.0f * un[i] - 1.0f) * 8.0f;
    atomicAdd(&hist16[mapf(an) >> 16], 1u);
}

// ---------- K10: global top-256 oldest select ----------
__global__ __launch_bounds__(1024) void age_select(const float* __restrict__ out_age,
                                                   const float* __restrict__ un,
                                                   const unsigned* __restrict__ hist16,
                                                   unsigned* __restrict__ oldest) {
    __shared__ unsigned long long candl[4096];
    __shared__ unsigned chunk[1024];
    __shared__ unsigned tbin, ccnt;
    int tid = threadIdx.x;
    unsigned s = 0;
    for (int b = 0; b < 64; ++b) s += hist16[tid * 64 + b];
    chunk[tid] = s;
    if (tid == 0) ccnt = 0;
    __syncthreads();
    if (tid == 0) {
        unsigned cum = 0; int c;
        for (c = 1023; c >= 0; --c) {
            if (cum + chunk[c] >= TOPK) break;
            cum += chunk[c];
        }
        if (c < 0) c = 0;
        unsigned t = (unsigned)c * 64;
        for (int b = 63; b >= 0; --b) {
            cum += hist16[c * 64 + b];
            if (cum >= TOPK) { t = (unsigned)(c * 64 + b); break; }
        }
        tbin = t;
    }
    __syncthreads();
    unsigned t = tbin;
    for (int i = tid; i < MEM; i += 1024) {
        float an = out_age[i] + (2.0f * un[i] - 1.0f) * 8.0f;
        unsigned u = mapf(an);
        if ((u >> 16) >= t) {
            unsigned p = atomicAdd(&ccnt, 1u);
            if (p < 4096) candl[p] = ((unsigned long long)(~u) << 32) | (unsigned)i;
        }
    }
    __syncthreads();
    unsigned n = ccnt > 4096 ? 4096 : ccnt;
    for (int i = tid; i < 4096; i += 1024) if (i >= (int)n) candl[i] = 0xFFFFFFFFFFFFFFFFull;
    __syncthreads();
    bitonic4096(candl);
    if (tid < TOPK) oldest[tid] = (unsigned)(candl[tid] & 0xFFFFFFFFu);
}

// ---------- K11: rank scan + dest + deterministic loss reduction ----------
__global__ __launch_bounds__(256) void rank_kernel(const int* __restrict__ correctw,
                                                   const unsigned* __restrict__ oldest,
                                                   const float* __restrict__ lossarr,
                                                   int* __restrict__ destw,
                                                   float* __restrict__ out_loss) {
    __shared__ int sc[256];
    __shared__ float lf[256];
    int tid = threadIdx.x;
    int inc = correctw[tid] ? 0 : 1;
    sc[tid] = inc; lf[tid] = lossarr[tid];
    __syncthreads();
    for (int off = 1; off < 256; off <<= 1) {
        int v = (tid >= off) ? sc[tid - off] : 0;
        __syncthreads();
        sc[tid] += v;
        __syncthreads();
    }
    int rank = sc[tid] - 1;
    destw[tid] = inc ? (int)oldest[rank < 0 ? 0 : rank] : MEM;
    for (int s = 128; s > 0; s >>= 1) { if (tid < s) lf[tid] += lf[tid + s]; __syncthreads(); }
    if (tid == 0) *out_loss = lf[0] * (1.0f / 256.0f);
}

// ---------- K12: incorrect-row scatter (dest indices are distinct) ----------
__global__ void incorrect_update(const float* __restrict__ q, const int* __restrict__ y,
                                 const int* __restrict__ destw,
                                 float* __restrict__ out_keys, int* __restrict__ out_vals,
                                 float* __restrict__ out_age) {
    int r = blockIdx.x, tid = threadIdx.x;
    int d = destw[r];
    if (d >= MEM) return;
    out_keys[(size_t)d * KD + tid] = q[r * KD + tid];
    if (tid == 0) { out_vals[d] = y[r]; out_age[d] = 0.0f; }
}

extern "C" void kernel_launch(void* const* d_in, const int* in_sizes, int n_in,
                              void* d_out, int out_size, void* d_ws, size_t ws_size,
                              hipStream_t stream) {
    const float* x      = (const float*)d_in[0];
    const int*   y      = (const int*)d_in[1];
    const float* W      = (const float*)d_in[2];
    const float* b      = (const float*)d_in[3];
    const float* keys   = (const float*)d_in[4];
    const int*   values = (const int*)d_in[5];
    const float* age    = (const float*)d_in[6];
    const float* unoise = (const float*)d_in[7];

    // outputs, concatenated flat in return order
    float* out      = (float*)d_out;
    int*   out_yhat = (int*)d_out;              // 256
    float* out_soft = out + 256;                // 65536
    float* out_loss = out + 65792;              // 1
    float* out_keys = out + 65793;              // 67108864
    int*   out_vals = (int*)d_out + 67174657;   // 262144
    float* out_age  = out + 67436801;           // 262144

    // scores scratch aliases the keys-output region (identical element count),
    // consumed by select kernels before copy_keys overwrites it.
    float* scores = out_keys;

    // workspace layout
    char* w = (char*)d_ws;
    float*    q        = (float*)w;                       // 256 KB
    unsigned* candcnt  = (unsigned*)(w + 262144);         // 1 KB
    unsigned* thresh   = (unsigned*)(w + 263168);         // 1 KB
    unsigned* yhatidx  = (unsigned*)(w + 264192);         // 1 KB
    int*      correctw = (int*)(w + 265216);              // 1 KB
    unsigned* oldest   = (unsigned*)(w + 266240);         // 1 KB
    int*      destw    = (int*)(w + 267264);              // 1 KB
    float*    lossarr  = (float*)(w + 268288);            // 1 KB
    unsigned* hist16   = (unsigned*)(w + 269312);         // 256 KB
    unsigned* hist12   = (unsigned*)(w + 532480);         // 4 MB
    unsigned long long* cand = (unsigned long long*)(w + 4726784); // 8 MB

    (void)hipMemsetAsync(candcnt, 0, BATCH * sizeof(unsigned), stream);
    (void)hipMemsetAsync(hist16, 0, 65536 * sizeof(unsigned), stream);

    query_kernel<<<BATCH, 256, 0, stream>>>(x, W, b, q);
    scores_kernel<<<MEM / (16 * TPB), 512, 0, stream>>>(q, keys, scores);
    hist12_kernel<<<BATCH, 1024, 0, stream>>>(scores, hist12);
    thresh_kernel<<<1, BATCH, 0, stream>>>(hist12, thresh);
    gather_kernel<<<BATCH, 1024, 0, stream>>>(scores, thresh, cand, candcnt);
    topk_kernel<<<BATCH, 1024, 0, stream>>>(cand, candcnt, values, y, out_soft,
                                            out_yhat, lossarr, yhatidx, correctw);
    copy_keys<<<8192, 256, 0, stream>>>(keys, out_keys);
    copy_vals_age<<<MEM / 256, 256, 0, stream>>>(values, age, out_vals, out_age);
    correct_update<<<1, 256, 0, stream>>>(keys, q, yhatidx, correctw, out_keys, out_age);
    age_hist<<<MEM / 256, 256, 0, stream>>>(out_age, unoise, hist16);
    age_select<<<1, 1024, 0, stream>>>(out_age, unoise, hist16, oldest);
    rank_kernel<<<1, 256, 0, stream>>>(correctw, oldest, lossarr, destw, out_loss);
    incorrect_update<<<BATCH, 256, 0, stream>>>(q, y, destw, out_keys, out_vals, out_age);
}